// SDTs_39487929319776
// MI455X (gfx1250) — compile-verified
//
#include <hip/hip_runtime.h>
#include <hip/hip_bf16.h>

// ---------------------------------------------------------------------------
// Soft Decision Tree ensemble loss, fused for gfx1250 (MI455X).
//   T=100 trees, N=8192 samples, N_X=512 features, INNER=31, LEAFS=32.
//   Main kernel: fp32 WMMA (v_wmma_f32_16x16x4_f32) GEMM fused with sigmoid,
//   tree path-probability products, leaf->phi dot, and regularizer partials.
//   W is staged in LDS k-pair interleaved (+96-dword row pitch) so each B
//   fragment is one conflict-free ds_load_b64 straight into the WMMA operand.
// ---------------------------------------------------------------------------

typedef float v2f __attribute__((ext_vector_type(2)));
typedef float v8f __attribute__((ext_vector_type(8)));

#define SDT_T      100
#define SDT_N      8192
#define SDT_NX     512
#define SDT_INNER  31
#define SDT_LEAFS  32
#define SDT_RBLKS  32          // 8192 / 256 rows per block
#define SDT_WPITCH 96          // dwords per k-pair row in LDS (bank-split pad)

// ---------------- workspace layout (floats) ----------------
// [0] meanY  [1] cost_wr  [2] sumW2  [3..15] pad
// [16 .. 16+T*N)                      : out[t][n]
// [po .. po+T*32*31)                  : psum  (sum_n inner_prob)
// [pa .. pa+T*32*31)                  : psumA (sum_n inner_prob*A)
#define WS_OUT_OFF   16
#define WS_PSUM_OFF  (WS_OUT_OFF + SDT_T * SDT_N)
#define WS_PSUMA_OFF (WS_PSUM_OFF + SDT_T * SDT_RBLKS * SDT_INNER)

__global__ __launch_bounds__(64) void sdt_init(float* acc) {
    if (threadIdx.x < 4) acc[threadIdx.x] = 0.0f;
}

__global__ __launch_bounds__(256) void sdt_meany(const float* __restrict__ Y,
                                                 float* __restrict__ meanY) {
    __shared__ float red[256];
    int tid = threadIdx.x;
    float s = 0.0f;
    for (int i = tid; i < SDT_N; i += 256) s += Y[i];
    red[tid] = s;
    __syncthreads();
    for (int w = 128; w > 0; w >>= 1) {
        if (tid < w) red[tid] += red[tid + w];
        __syncthreads();
    }
    if (tid == 0) meanY[0] = red[0] * (1.0f / (float)SDT_N);
}

// ---------------------------------------------------------------------------
// Fused GEMM (WMMA f32 16x16x4) + sigmoid + soft-tree epilogue.
// gridDim = (32 row-blocks, 100 trees), blockDim = 256 (8 waves, wave32).
// Each wave computes a 32x32 tile of logits = X(32x512) * W_t(512x32pad).
// ---------------------------------------------------------------------------
__global__ __launch_bounds__(256) void sdt_gemm_tree(
    const float* __restrict__ X, const float* __restrict__ W,
    const float* __restrict__ bvec, const float* __restrict__ phi,
    float* __restrict__ outArr, float* __restrict__ psum,
    float* __restrict__ psumA) {
    const int t = blockIdx.y;
    const int rowBlk = blockIdx.x;
    const int rowBase = rowBlk * 256;
    const int tid = threadIdx.x;
    const int wave = tid >> 5;
    const int lane = tid & 31;
    const int lo = lane & 15;
    const int hi = lane >> 4;

    // 64-K slab of W_t, k-pair interleaved: element (k,c) lives at
    // (k/2)*SDT_WPITCH + c*2 + (k&1). Column 31 is zero-padded.
    __shared__ float Wlds[32 * SDT_WPITCH];
    __shared__ float Alds[8][32 * 33];     // per-wave sigmoid(A) tile, padded
    __shared__ float phis[SDT_LEAFS];
    __shared__ float bs[32];
    __shared__ float sacc[2 * SDT_INNER];  // [0..30] sum(ip), [31..61] sum(ip*A)

    if (tid < 2 * SDT_INNER) sacc[tid] = 0.0f;
    if (tid >= 64 && tid < 96) phis[tid - 64] = phi[t * SDT_LEAFS + (tid - 64)];
    if (tid >= 96 && tid < 128) {
        int i = tid - 96;
        bs[i] = (i < SDT_INNER) ? bvec[t * SDT_INNER + i] : 0.0f;
    }

    v8f acc[2][2] = {};
    const float* Wt = W + (size_t)t * SDT_NX * SDT_INNER;
    const int row0 = rowBase + wave * 32 + lo;
    const float* xr0 = X + (size_t)row0 * SDT_NX;
    const float* xr1 = xr0 + (size_t)16 * SDT_NX;

    for (int ks = 0; ks < SDT_NX; ks += 64) {
        __syncthreads();
        // stage 64x32 slab of W_t into LDS, k-pair interleaved
        for (int i = tid; i < 64 * 32; i += 256) {
            int kr = i >> 5;
            int c = i & 31;
            float v = (c < SDT_INNER) ? Wt[(size_t)(ks + kr) * SDT_INNER + c]
                                      : 0.0f;
            Wlds[(kr >> 1) * SDT_WPITCH + c * 2 + (kr & 1)] = v;
        }
        __syncthreads();
#pragma unroll
        for (int kk = 0; kk < 64; kk += 4) {
            const int kb = kk + 2 * hi;  // lanes 0-15: K=kk, lanes 16-31: K=kk+2
            const int kp = kb >> 1;      // k-pair row in LDS
            // A fragments (16x4 f32 layout): float2 per lane from global (L2)
            float2 av0 = *(const float2*)(xr0 + ks + kb);
            float2 av1 = *(const float2*)(xr1 + ks + kb);
            v2f a0 = {av0.x, av0.y};
            v2f a1 = {av1.x, av1.y};
            // B fragments: one contiguous ds_load_b64 each, bank-conflict-free
            float2 bv0 = *(const float2*)&Wlds[kp * SDT_WPITCH + lo * 2];
            float2 bv1 = *(const float2*)&Wlds[kp * SDT_WPITCH + 32 + lo * 2];
            v2f b0 = {bv0.x, bv0.y};
            v2f b1 = {bv1.x, bv1.y};
            acc[0][0] = __builtin_amdgcn_wmma_f32_16x16x4_f32(
                false, a0, false, b0, (short)0, acc[0][0], false, false);
            acc[0][1] = __builtin_amdgcn_wmma_f32_16x16x4_f32(
                false, a0, false, b1, (short)0, acc[0][1], false, false);
            acc[1][0] = __builtin_amdgcn_wmma_f32_16x16x4_f32(
                false, a1, false, b0, (short)0, acc[1][0], false, false);
            acc[1][1] = __builtin_amdgcn_wmma_f32_16x16x4_f32(
                false, a1, false, b1, (short)0, acc[1][1], false, false);
        }
    }

    // Epilogue: bias + sigmoid, stage A tile to LDS (row-padded to 33 floats)
#pragma unroll
    for (int r = 0; r < 2; ++r) {
#pragma unroll
        for (int c = 0; c < 2; ++c) {
#pragma unroll
            for (int j = 0; j < 8; ++j) {
                int lr = r * 16 + j + 8 * hi;  // local row within 32-row tile
                int col = c * 16 + lo;
                float logit = acc[r][c][j] + bs[col];
                Alds[wave][lr * 33 + col] = 1.0f / (1.0f + __expf(-logit));
            }
        }
    }
    __syncthreads();

    // Per-lane soft decision tree: one sample per lane (32 samples per wave)
    const int n = rowBase + wave * 32 + lane;
    const float* Av = &Alds[wave][lane * 33];
    float mu[SDT_LEAFS];
    mu[0] = 1.0f;
#pragma unroll
    for (int d = 1; d < 6; ++d) {
        const int half = 1 << (d - 1);
        const int off = half - 1;
#pragma unroll
        for (int i = half - 1; i >= 0; --i) {
            float p = mu[i];            // inner_prob of node (off+i)
            float Af = Av[off + i];
            atomicAdd(&sacc[off + i], p);
            atomicAdd(&sacc[SDT_INNER + off + i], p * Af);
            float pr = p * Af;
            mu[2 * i + 1] = pr;
            mu[2 * i] = p - pr;
        }
    }
    float o = 0.0f;
#pragma unroll
    for (int l = 0; l < SDT_LEAFS; ++l) o += mu[l] * phis[l];
    outArr[(size_t)t * SDT_N + n] = o;

    __syncthreads();
    if (tid < SDT_INNER) {
        size_t base = ((size_t)t * SDT_RBLKS + rowBlk) * SDT_INNER + tid;
        psum[base] = sacc[tid];
        psumA[base] = sacc[SDT_INNER + tid];
    }
}

// Per-sample sequential scan over trees (cumsum -> exp loss -> residual^2)
__global__ __launch_bounds__(256) void sdt_loss(const float* __restrict__ outArr,
                                                const float* __restrict__ Y,
                                                const float* __restrict__ meanY,
                                                float* __restrict__ cost) {
    const int n = blockIdx.x * 256 + threadIdx.x;  // 32 blocks x 256 = 8192
    const float y = Y[n];
    float os = meanY[0];
    float c = 0.0f;
    for (int t = 0; t < SDT_T; ++t) {
        float o = outArr[(size_t)t * SDT_N + n];
        os += 0.1f * o;
        float r = __expf(-os * y) * y;   // residual = exp(-out_sum*y)*y
        float d = r - o;
        c += d * d;
    }
    __shared__ float red[256];
    red[threadIdx.x] = c;
    __syncthreads();
    for (int w = 128; w > 0; w >>= 1) {
        if (threadIdx.x < w) red[threadIdx.x] += red[threadIdx.x + w];
        __syncthreads();
    }
    if (threadIdx.x == 0) atomicAdd(cost, red[0]);
}

__global__ __launch_bounds__(256) void sdt_sumw2(const float* __restrict__ W,
                                                 float* __restrict__ sumW2) {
    const size_t total = (size_t)SDT_T * SDT_NX * SDT_INNER;
    float s = 0.0f;
    for (size_t i = blockIdx.x * 256 + threadIdx.x; i < total; i += 256 * 256) {
        float w = W[i];
        s += w * w;
    }
    __shared__ float red[256];
    red[threadIdx.x] = s;
    __syncthreads();
    for (int w = 128; w > 0; w >>= 1) {
        if (threadIdx.x < w) red[threadIdx.x] += red[threadIdx.x + w];
        __syncthreads();
    }
    if (threadIdx.x == 0) atomicAdd(sumW2, red[0]);
}

// Regularizer from partials + final scalar combine
__global__ __launch_bounds__(128) void sdt_final(const float* __restrict__ psum,
                                                 const float* __restrict__ psumA,
                                                 const float* __restrict__ cost,
                                                 const float* __restrict__ sumW2,
                                                 float* __restrict__ outp) {
    const int tid = threadIdx.x;
    float reg = 0.0f;
    for (int idx = tid; idx < SDT_T * SDT_INNER; idx += 128) {
        int t = idx / SDT_INNER;
        int k = idx % SDT_INNER;
        float sp = 0.0f, spa = 0.0f;
        for (int b2 = 0; b2 < SDT_RBLKS; ++b2) {
            size_t base = ((size_t)t * SDT_RBLKS + b2) * SDT_INNER + k;
            sp += psum[base];
            spa += psumA[base];
        }
        float a = spa / sp;
        float df = (float)(31 - __clz(k + 1));       // floor(log2(k+1))
        float w = __expf(-df) * a;                   // (1-decay)*a
        float scale = exp2f(-df);
        reg += -0.1f * scale * (0.5f * __logf(w) + 0.5f * __logf(1.0f - w));
    }
    __shared__ float red[128];
    red[tid] = reg;
    __syncthreads();
    for (int w = 64; w > 0; w >>= 1) {
        if (tid < w) red[tid] += red[tid + w];
        __syncthreads();
    }
    if (tid == 0) outp[0] = cost[0] + red[0] + 0.005f * sumW2[0];
}

extern "C" void kernel_launch(void* const* d_in, const int* in_sizes, int n_in,
                              void* d_out, int out_size, void* d_ws, size_t ws_size,
                              hipStream_t stream) {
    const float* X    = (const float*)d_in[0];  // (8192, 512)
    const float* Y    = (const float*)d_in[1];  // (8192,)
    const float* W    = (const float*)d_in[2];  // (100, 512, 31)
    const float* bvec = (const float*)d_in[3];  // (100, 1, 31)
    const float* phi  = (const float*)d_in[4];  // (100, 32, 1)
    float* out = (float*)d_out;

    float* ws      = (float*)d_ws;
    float* wsOut   = ws + WS_OUT_OFF;
    float* wsPsum  = ws + WS_PSUM_OFF;
    float* wsPsumA = ws + WS_PSUMA_OFF;

    sdt_init<<<1, 64, 0, stream>>>(ws);
    sdt_meany<<<1, 256, 0, stream>>>(Y, ws /*meanY at [0]*/);
    sdt_gemm_tree<<<dim3(SDT_RBLKS, SDT_T), 256, 0, stream>>>(
        X, W, bvec, phi, wsOut, wsPsum, wsPsumA);
    sdt_loss<<<SDT_N / 256, 256, 0, stream>>>(wsOut, Y, ws, ws + 1);
    sdt_sumw2<<<256, 256, 0, stream>>>(W, ws + 2);
    sdt_final<<<1, 128, 0, stream>>>(wsPsum, wsPsumA, ws + 1, ws + 2, out);
}